// GCNConv_12927851561521
// MI455X (gfx1250) — compile-verified
//
#include <hip/hip_runtime.h>
#include <math.h>

typedef __attribute__((ext_vector_type(2))) float v2f;
typedef __attribute__((ext_vector_type(8))) float v8f;

#define BB 8
#define PP 4
#define NN 2048
#define UU 64
#define CCH 16              // UU / PP, matches WMMA M
#define SLICES (BB * PP)    // 32
#define EPSF 1e-7f

// ---------------------------------------------------------------------------
// Kernel 1: deg[s,m] = sum_i Ans[s,i,m];  dhat = 1/(sqrt(deg)+eps)
// ---------------------------------------------------------------------------
__global__ void k_degree(const float* __restrict__ Ans,
                         float* __restrict__ deg,
                         float* __restrict__ dhat) {
    const int t = blockIdx.x * blockDim.x + threadIdx.x;   // 0 .. SLICES*NN-1
    const int s = t / NN;
    const int m = t - s * NN;
    const float* base = Ans + (size_t)s * NN * NN + m;
    float sum = 0.f;
    #pragma unroll 8
    for (int i = 0; i < NN; ++i) sum += base[(size_t)i * NN];
    deg[t]  = sum;
    dhat[t] = 1.0f / (sqrtf(sum) + EPSF);
}

// ---------------------------------------------------------------------------
// Kernel 2: h[b,n,u] = sum_w X[b,n,w] * W[w,u];  G[s,c,n] = h * dhat[s,n]
// ---------------------------------------------------------------------------
__global__ void k_gmat(const float* __restrict__ X,
                       const float* __restrict__ W,
                       const float* __restrict__ dhat,
                       float* __restrict__ G) {
    const int t = blockIdx.x * blockDim.x + threadIdx.x;   // 0 .. B*NN*UU-1
    const int u    = t % UU;
    const int nidx = (t / UU) % NN;
    const int b    = t / (UU * NN);
    const float* xrow = X + ((size_t)b * NN + nidx) * UU;
    float h = 0.f;
    #pragma unroll
    for (int w = 0; w < UU; ++w) h += xrow[w] * W[w * UU + u];
    const int p = u / CCH, c = u % CCH;
    const int s = b * PP + p;
    G[((size_t)s * CCH + c) * NN + nidx] = h * dhat[s * NN + nidx];
}

// ---------------------------------------------------------------------------
// Kernel 3: one wave per (slice, PAIR of 16-wide m tiles) => 32-col stripe.
// Full 128B cacheline per Ans row per wave; one shared A-fragment feeds two
// independent v_wmma_f32_16x16x4_f32 accumulator chains. Software-pipelined:
// next step's fragments are loaded before this step's WMMAs.
// ---------------------------------------------------------------------------
__global__ void __launch_bounds__(256) k_wmma(const float* __restrict__ Ans,
                                              const float* __restrict__ G,
                                              const float* __restrict__ deg,
                                              const float* __restrict__ dhat,
                                              const float* __restrict__ bias,
                                              float* __restrict__ out) {
    const int lane = threadIdx.x & 31;
    const int wid  = (blockIdx.x * blockDim.x + threadIdx.x) >> 5; // 0..2047
    const int s    = wid >> 6;         // slice (0..31)
    const int tp   = wid & 63;         // tile pair (0..63)
    const int m0   = tp * 32;
    const int half = lane >> 4;        // 0 or 1
    const int l15  = lane & 15;
    const int mm0  = m0 + l15;         // tile-0 column
    const int mm1  = mm0 + 16;         // tile-1 column

    const float* As   = Ans + (size_t)s * NN * NN;
    const float* Gs   = G   + (size_t)s * CCH * NN;
    const float* grow = Gs  + (size_t)l15 * NN;
    const int    koff = 2 * half;

    const float* ap  = grow + koff;                       // A-frag base (k)
    const float* bp0 = As + (size_t)koff * NN + mm0;      // B-frag tile0 base
    const float* bp1 = As + (size_t)koff * NN + mm1;      // B-frag tile1 base

    // ---- prologue: load fragments for k0 = 0 ----
    v2f a_c = *(const v2f*)ap;
    v2f b0_c, b1_c;
    b0_c.x = bp0[0];  b0_c.y = bp0[NN];
    b1_c.x = bp1[0];  b1_c.y = bp1[NN];

    v8f acc0 = {}, acc1 = {};

    // ---- steady state: issue loads for k0+4 before WMMAs of k0 ----
    #pragma unroll 4
    for (int k0 = 0; k0 < NN - 4; k0 += 4) {
        v2f a_n = *(const v2f*)(ap + (k0 + 4));
        const float* nb0 = bp0 + (size_t)(k0 + 4) * NN;
        const float* nb1 = bp1 + (size_t)(k0 + 4) * NN;
        v2f b0_n, b1_n;
        b0_n.x = nb0[0];  b0_n.y = nb0[NN];
        b1_n.x = nb1[0];  b1_n.y = nb1[NN];
        if (k0 + 36 < NN)   // uniform branch; EXEC unchanged
            __builtin_prefetch(bp0 + (size_t)(k0 + 32) * NN, 0, 3);
        acc0 = __builtin_amdgcn_wmma_f32_16x16x4_f32(
            false, a_c, false, b0_c, (short)0, acc0, false, false);
        acc1 = __builtin_amdgcn_wmma_f32_16x16x4_f32(
            false, a_c, false, b1_c, (short)0, acc1, false, false);
        a_c = a_n;  b0_c = b0_n;  b1_c = b1_n;
    }
    // ---- tail: last K block ----
    acc0 = __builtin_amdgcn_wmma_f32_16x16x4_f32(
        false, a_c, false, b0_c, (short)0, acc0, false, false);
    acc1 = __builtin_amdgcn_wmma_f32_16x16x4_f32(
        false, a_c, false, b1_c, (short)0, acc1, false, false);

    // ---- epilogue: out[b, m, p*16+c] = dhat[m]*(G[c,m]*deg[m] - S) + bias ----
    const int b = s >> 2;              // s / PP
    const int p = s & 3;               // s % PP
    const float dm0 = dhat[s * NN + mm0], dg0 = deg[s * NN + mm0];
    const float dm1 = dhat[s * NN + mm1], dg1 = deg[s * NN + mm1];
    float* orow0 = out + ((size_t)b * NN + mm0) * UU + p * CCH;
    float* orow1 = out + ((size_t)b * NN + mm1) * UU + p * CCH;
    #pragma unroll
    for (int r = 0; r < 8; ++r) {
        const int c  = r + 8 * half;
        const float bv = bias[p * CCH + c];
        orow0[c] = dm0 * (Gs[(size_t)c * NN + mm0] * dg0 - acc0[r]) + bv;
        orow1[c] = dm1 * (Gs[(size_t)c * NN + mm1] * dg1 - acc1[r]) + bv;
    }
}

// ---------------------------------------------------------------------------
extern "C" void kernel_launch(void* const* d_in, const int* in_sizes, int n_in,
                              void* d_out, int out_size, void* d_ws, size_t ws_size,
                              hipStream_t stream) {
    const float* Ans  = (const float*)d_in[0];  // [B,P,n,n]
    const float* X    = (const float*)d_in[1];  // [B,n,U]
    const float* W    = (const float*)d_in[2];  // [U,U]
    const float* bias = (const float*)d_in[3];  // [U]
    float* out = (float*)d_out;                 // [B,n,U]

    float* deg  = (float*)d_ws;                 // SLICES*NN
    float* dhat = deg  + (size_t)SLICES * NN;   // SLICES*NN
    float* G    = dhat + (size_t)SLICES * NN;   // SLICES*CCH*NN

    // 1) column sums + dhat : 65536 threads
    k_degree<<<(SLICES * NN) / 256, 256, 0, stream>>>(Ans, deg, dhat);
    // 2) h = X@W folded with dhat into G : 1M threads
    k_gmat<<<(BB * NN * UU) / 256, 256, 0, stream>>>(X, W, dhat, G);
    // 3) WMMA main pass: 32 slices * 64 tile-pairs = 2048 waves, 8 waves/block
    k_wmma<<<(SLICES * 64 * 32) / 256, 256, 0, stream>>>(
        Ans, G, deg, dhat, bias, out);
}